// RegionLoss_75496935129670
// MI455X (gfx1250) — compile-verified
//
#include <hip/hip_runtime.h>
#include <math.h>

// ---------------------------------------------------------------------------
// YOLO region loss, fused single pass. Shapes fixed by setup_inputs():
//   x: (64, 125, 52, 52) f32, label: (64, 150) f32, anchors: (5,2) f32
// Memory-bound: only read the 5 box/conf channels per cell everywhere
// (17.3 MB); class channels (80% of x) only at matched GT cells (~30/image).
// Reductions use V_WMMA_F32_16X16X4_F32 with A=ones (D rows = column sums,
// layout-agnostic total), per CDNA5 WMMA spec.
// ---------------------------------------------------------------------------

#define RB 64
#define RA 5
#define RH 52
#define RW 52
#define RM 30
#define RC 20
#define RHW (RH * RW)              // 2704
#define RCELLS (RA * RHW)          // 13520
#define RCH 125
#define BLOCK 256
#define GRIDX ((RCELLS + BLOCK - 1) / BLOCK)   // 53
#define NPART (GRIDX * RB)                     // 3392 (= 53 * 64, exact)

typedef __attribute__((ext_vector_type(2))) float v2f;
typedef __attribute__((ext_vector_type(8))) float v8f;

__device__ __forceinline__ float sigmoidf(float v) {
    return 1.0f / (1.0f + expf(-v));
}

// Sum 64 floats (2 per lane of a wave32) via one WMMA accumulate step.
// A = all ones (16x4) => D[i][j] = sum_k B[k][j]; independent of B layout,
// sum of any full D row equals the total of all 64 B elements.
__device__ __forceinline__ v8f wmma_accum64(v8f acc, float b0, float b1) {
    v2f one = {1.0f, 1.0f};
    v2f bv;
    bv.x = b0;
    bv.y = b1;
    return __builtin_amdgcn_wmma_f32_16x16x4_f32(
        /*neg_a=*/false, one, /*neg_b=*/false, bv,
        /*c_mod=*/(short)0, acc, /*reuse_a=*/false, /*reuse_b=*/false);
}

// After accumulation: vgpr0 holds row0 (lanes 0-15) and row8 (lanes 16-31);
// all rows identical => sum over 32 lanes = 2 * total.
__device__ __forceinline__ float wmma_extract_total(v8f acc) {
    float v = acc[0];
    v += __shfl_xor(v, 16, 32);
    v += __shfl_xor(v, 8, 32);
    v += __shfl_xor(v, 4, 32);
    v += __shfl_xor(v, 2, 32);
    v += __shfl_xor(v, 1, 32);
    return 0.5f * v;
}

__global__ __launch_bounds__(BLOCK) void region_loss_main(
    const float* __restrict__ x, const float* __restrict__ label,
    const float* __restrict__ anchors, float* __restrict__ partial)
{
    __shared__ float s_aw[RA], s_ah[RA];
    __shared__ int   s_valid[RM], s_gxi[RM], s_gyi[RM], s_ba[RM], s_cls[RM];
    __shared__ float s_x1[RM], s_y1[RM], s_x2[RM], s_y2[RM], s_area[RM];
    __shared__ float s_gtx[RM], s_gty[RM], s_gtw[RM], s_gth[RM], s_cval[RM];
    __shared__ float s_ux1[RM], s_uy1[RM], s_ux2[RM], s_uy2[RM], s_uarea[RM];
    __shared__ float s_loss[BLOCK];

    const int b   = blockIdx.y;
    const int tid = threadIdx.x;

    if (tid < RA) {
        s_aw[tid] = anchors[2 * tid];
        s_ah[tid] = anchors[2 * tid + 1];
    }
    if (tid < RM) {
        const float* t = label + (size_t)b * (RM * 5) + tid * 5;
        float txn = t[0], tyn = t[1], twn = t[2], thn = t[3], clsf = t[4];
        int valid = (txn != 0.0f);
        s_valid[tid] = valid;
        float cx = txn * RW, cy = tyn * RH;
        float gw = twn * RW, gh = thn * RH;
        s_x1[tid] = cx - 0.5f * gw;  s_y1[tid] = cy - 0.5f * gh;
        s_x2[tid] = cx + 0.5f * gw;  s_y2[tid] = cy + 0.5f * gh;
        s_area[tid] = gw * gh;
        int gxi = (int)cx, gyi = (int)cy;
        s_gxi[tid] = gxi; s_gyi[tid] = gyi;
        // best anchor (first max wins, like jnp.argmax)
        float besti = -1.0f; int ba = 0;
        for (int k = 0; k < RA; ++k) {
            float aw = anchors[2 * k], ah = anchors[2 * k + 1];
            float inter = fminf(aw, gw) * fminf(ah, gh);
            float iou = inter / (aw * ah + gw * gh - inter);
            if (iou > besti) { besti = iou; ba = k; }
        }
        s_ba[tid]  = ba;
        s_gtx[tid] = cx - (float)gxi;
        s_gty[tid] = cy - (float)gyi;
        s_gtw[tid] = valid ? logf(gw / anchors[2 * ba])     : 0.0f;
        s_gth[tid] = valid ? logf(gh / anchors[2 * ba + 1]) : 0.0f;
        s_cval[tid] = 2.0f - twn * thn;     // COORD_SCALE * (2 - w*h)
        s_cls[tid]  = (int)clsf;
        s_ux1[tid] = txn - 0.5f * twn;  s_uy1[tid] = tyn - 0.5f * thn;
        s_ux2[tid] = txn + 0.5f * twn;  s_uy2[tid] = tyn + 0.5f * thn;
        s_uarea[tid] = twn * thn;
    }
    __syncthreads();

    float contrib = 0.0f;
    const int cell = blockIdx.x * BLOCK + tid;
    if (cell < RCELLS) {
        const int a  = cell / RHW;
        const int hw = cell - a * RHW;
        const int h  = hw / RW;
        const int w  = hw - h * RW;
        // x[b, k*A + a, h, w]; consecutive w => coalesced per channel plane
        const size_t base = (size_t)b * RCH * RHW + hw;
        float lx = x[base + (size_t)(0 * RA + a) * RHW];
        float ly = x[base + (size_t)(1 * RA + a) * RHW];
        float lw = x[base + (size_t)(2 * RA + a) * RHW];
        float lh = x[base + (size_t)(3 * RA + a) * RHW];
        float lc = x[base + (size_t)(4 * RA + a) * RHW];

        float sx   = sigmoidf(lx);
        float sy   = sigmoidf(ly);
        float conf = sigmoidf(lc);
        float aw = s_aw[a], ah = s_ah[a];
        float pw = expf(lw) * aw, ph = expf(lh) * ah;
        float pcx = (float)w + sx, pcy = (float)h + sy;
        float px1 = pcx - 0.5f * pw, py1 = pcy - 0.5f * ph;
        float px2 = pcx + 0.5f * pw, py2 = pcy + 0.5f * ph;
        float parea = pw * ph;

        float best_iou = -1.0f;
        int match = -1;                      // last-match-wins (scatter .set)
        for (int m = 0; m < RM; ++m) {
            if (!s_valid[m]) continue;
            float ix1 = fmaxf(px1, s_x1[m]), iy1 = fmaxf(py1, s_y1[m]);
            float ix2 = fminf(px2, s_x2[m]), iy2 = fminf(py2, s_y2[m]);
            float inter = fmaxf(ix2 - ix1, 0.0f) * fmaxf(iy2 - iy1, 0.0f);
            float iou = inter / (parea + s_area[m] - inter);
            best_iou = fmaxf(best_iou, iou);
            if (s_ba[m] == a && s_gyi[m] == h && s_gxi[m] == w) match = m;
        }

        float cmask = (best_iou > 0.6f) ? 0.0f : 1.0f;   // NOOBJECT_SCALE
        float cm = 0.01f, tx = 0.5f, ty = 0.5f, tw = 0.0f, th = 0.0f;
        float tconf = 0.0f;
        int tcls = -1;
        if (match >= 0) {
            cm = s_cval[match]; cmask = 5.0f;            // OBJECT_SCALE
            tx = s_gtx[match];  ty = s_gty[match];
            tw = s_gtw[match];  th = s_gth[match];
            tcls = s_cls[match];
            // diag IoU in normalized units: pred corners / 52
            const float inv = 1.0f / 52.0f;
            float bx1 = px1 * inv, by1 = py1 * inv;
            float bx2 = px2 * inv, by2 = py2 * inv;
            float ix1 = fmaxf(bx1, s_ux1[match]), iy1 = fmaxf(by1, s_uy1[match]);
            float ix2 = fminf(bx2, s_ux2[match]), iy2 = fminf(by2, s_uy2[match]);
            float inter = fmaxf(ix2 - ix1, 0.0f) * fmaxf(iy2 - iy1, 0.0f);
            tconf = inter / (s_uarea[match] + parea * inv * inv - inter);
        }

        float l = cm * ((sx - tx) * (sx - tx) + (sy - ty) * (sy - ty) +
                        (lw - tw) * (lw - tw) + (lh - th) * (lh - th))
                + cmask * (conf - tconf) * (conf - tconf);
        contrib = l * (1.0f / (2.0f * RB));

        if (tcls >= 0) {
            // class logits for anchor slot a: channels 25 + 20*a + c
            const float* cl = x + ((size_t)b * RCH + 25 + RC * a) * RHW + hw;
            float mx = cl[0];
            for (int c = 1; c < RC; ++c) mx = fmaxf(mx, cl[(size_t)c * RHW]);
            float s = 0.0f;
            for (int c = 0; c < RC; ++c) s += expf(cl[(size_t)c * RHW] - mx);
            float lse = mx + logf(s);
            float nll = lse - cl[(size_t)tcls * RHW];
            contrib += nll * (1.0f / RB);
        }
    }
    s_loss[tid] = contrib;
    __syncthreads();

    // Block reduction on the matrix pipe: wave 0 only (EXEC all ones for
    // threads 0..31 => legal WMMA), 4 chained WMMAs consume 256 floats.
    if (tid < 32) {
        v8f acc = {};
        for (int it = 0; it < BLOCK / 64; ++it) {
            acc = wmma_accum64(acc, s_loss[it * 64 + tid],
                                    s_loss[it * 64 + 32 + tid]);
        }
        float total = wmma_extract_total(acc);
        if (tid == 0)
            partial[(size_t)blockIdx.y * gridDim.x + blockIdx.x] = total;
    }
}

__global__ __launch_bounds__(32) void region_loss_finish(
    const float* __restrict__ partial, float* __restrict__ out)
{
    const int lane = threadIdx.x;   // single wave32, EXEC all ones
    v8f acc = {};
    for (int base = 0; base < NPART; base += 64) {
        acc = wmma_accum64(acc, partial[base + lane],
                                partial[base + 32 + lane]);
    }
    float total = wmma_extract_total(acc);
    if (lane == 0) out[0] = total;
}

extern "C" void kernel_launch(void* const* d_in, const int* in_sizes, int n_in,
                              void* d_out, int out_size, void* d_ws, size_t ws_size,
                              hipStream_t stream) {
    (void)in_sizes; (void)n_in; (void)out_size; (void)ws_size;
    const float* x       = (const float*)d_in[0];
    const float* label   = (const float*)d_in[1];
    const float* anchors = (const float*)d_in[2];
    float* partial = (float*)d_ws;            // NPART floats of scratch

    dim3 grid(GRIDX, RB);
    region_loss_main<<<grid, BLOCK, 0, stream>>>(x, label, anchors, partial);
    region_loss_finish<<<1, 32, 0, stream>>>(partial, (float*)d_out);
}